// GIN_1090921693628
// MI455X (gfx1250) — compile-verified
//
#include <hip/hip_runtime.h>
#include <hip/hip_bf16.h>

typedef float v2f __attribute__((ext_vector_type(2)));
typedef float v8f __attribute__((ext_vector_type(8)));

#define NODES 100000
#define EDGES 1600000
#define FIN   128
#define HDIM  64
#define GRAPHS 512
#define TDIM  10
#define BN_EPS 1e-5f

// ---------------------------------------------------------------------------
// Z[n x 64] = A[n x K] @ W[K x 64] + bias   (f32, exact, V_WMMA_F32_16X16X4_F32)
// block = 256 threads = 8 waves (wave32); each wave computes 16 rows x 64 cols.
// W staged in LDS pre-paired: Wl[p*64+c] = {W[2p][c], W[2p+1][c]} so each B
// fragment is one contiguous ds_load_b64. A fragment is one global_load_b64
// with clamped (not predicated) addressing.
// ---------------------------------------------------------------------------
__global__ void gemm_bias_wmma(const float* __restrict__ A,
                               const float* __restrict__ W,
                               const float* __restrict__ bias,
                               float* __restrict__ Z,
                               int n, int K) {
    __shared__ float2 Wl[(FIN / 2) * HDIM];       // up to 64 pairs x 64 cols = 32 KB
    for (int i = threadIdx.x; i < (K / 2) * HDIM; i += blockDim.x) {
        const int pr = i >> 6, c = i & 63;
        Wl[i] = make_float2(W[(2 * pr) * HDIM + c], W[(2 * pr + 1) * HDIM + c]);
    }
    __syncthreads();

    const int wave = threadIdx.x >> 5;            // wave32 on gfx1250
    const int lane = threadIdx.x & 31;
    const int halfsel = lane >> 4;                // 0: lanes 0-15, 1: lanes 16-31
    const int lr = lane & 15;
    const int rowBase = (blockIdx.x * 8 + wave) * 16;
    const int row = rowBase + lr;                 // A-fragment row for this lane
    // clamp address, never predicate: OOB lanes read row 0, their C rows are
    // never stored. Keeps EXEC all-ones (required for WMMA) and loads tight.
    const float* Arow = A + (size_t)(row < n ? row : 0) * (size_t)K;

    v8f c[4];
#pragma unroll
    for (int t = 0; t < 4; ++t) c[t] = (v8f){};

    for (int k = 0; k < K; k += 4) {
        const int pb = (k >> 1) + halfsel;        // K-pair index for this lane half
        const float2 av = *(const float2*)(Arow + 2 * pb);
        v2f a; a.x = av.x; a.y = av.y;            // A layout: VGPR0=K{0,2}, VGPR1=K{1,3}
        const float2* wp = Wl + pb * HDIM;
#pragma unroll
        for (int t = 0; t < 4; ++t) {             // 4 column tiles of 16
            const float2 bv = wp[lr + 16 * t];
            v2f b; b.x = bv.x; b.y = bv.y;
            c[t] = __builtin_amdgcn_wmma_f32_16x16x4_f32(
                false, a, false, b, (short)0, c[t], false, false);
        }
    }

    // C layout: VGPR i -> lanes0-15: M=i, lanes16-31: M=i+8; N = lane%16 (+16*tile)
    if (rowBase + 16 <= n) {                      // wave-uniform: full tile, no guards
#pragma unroll
        for (int t = 0; t < 4; ++t) {
            const int col = lr + 16 * t;
            const float bv = bias[col];
#pragma unroll
            for (int i = 0; i < 8; ++i) {
                const int r = rowBase + i + 8 * halfsel;
                Z[(size_t)r * HDIM + col] = c[t][i] + bv;
            }
        }
    } else {
#pragma unroll
        for (int t = 0; t < 4; ++t) {
            const int col = lr + 16 * t;
            const float bv = bias[col];
#pragma unroll
            for (int i = 0; i < 8; ++i) {
                const int r = rowBase + i + 8 * halfsel;
                if (r < n) Z[(size_t)r * HDIM + col] = c[t][i] + bv;
            }
        }
    }
}

// ---------------------------------------------------------------------------
// Column sum / sum-of-squares over Z[n x 64] -> stats[0..63]=sum, [64..127]=sumsq
// ---------------------------------------------------------------------------
__global__ void col_stats(const float* __restrict__ Z, float* __restrict__ stats, int n) {
    const int col = threadIdx.x & 63;
    const int rl  = threadIdx.x >> 6;             // 0..3
    float s = 0.0f, sq = 0.0f;
    for (int r = blockIdx.x * 4 + rl; r < n; r += gridDim.x * 4) {
        const float v = Z[(size_t)r * HDIM + col];
        s += v; sq += v * v;
    }
    __shared__ float ls[4][64], lq[4][64];
    ls[rl][col] = s; lq[rl][col] = sq;
    __syncthreads();
    if (rl == 0) {
        s  = ls[0][col] + ls[1][col] + ls[2][col] + ls[3][col];
        sq = lq[0][col] + lq[1][col] + lq[2][col] + lq[3][col];
        atomicAdd(&stats[col], s);
        atomicAdd(&stats[64 + col], sq);
    }
}

// ---------------------------------------------------------------------------
// H = relu((Z - mean) * rsqrt(var + eps) * g + be); optional dual store.
// float4-vectorized: one thread per 4 consecutive features.
// ---------------------------------------------------------------------------
__global__ void bn_relu(const float* __restrict__ Z, const float* __restrict__ stats,
                        const float* __restrict__ g, const float* __restrict__ be,
                        float* __restrict__ H, float* __restrict__ H2, int n) {
    const int i = blockIdx.x * blockDim.x + threadIdx.x;   // over n*16 float4s
    if (i >= n * 16) return;
    const int cb = (i & 15) * 4;
    const float invN = 1.0f / (float)n;
    const float4 z = *(const float4*)(Z + (size_t)i * 4);
    float4 y;
    {
        float m, v, r;
        m = stats[cb + 0] * invN; v = stats[64 + cb + 0] * invN - m * m; r = rsqrtf(v + BN_EPS);
        y.x = fmaxf((z.x - m) * r * g[cb + 0] + be[cb + 0], 0.0f);
        m = stats[cb + 1] * invN; v = stats[64 + cb + 1] * invN - m * m; r = rsqrtf(v + BN_EPS);
        y.y = fmaxf((z.y - m) * r * g[cb + 1] + be[cb + 1], 0.0f);
        m = stats[cb + 2] * invN; v = stats[64 + cb + 2] * invN - m * m; r = rsqrtf(v + BN_EPS);
        y.z = fmaxf((z.z - m) * r * g[cb + 2] + be[cb + 2], 0.0f);
        m = stats[cb + 3] * invN; v = stats[64 + cb + 3] * invN - m * m; r = rsqrtf(v + BN_EPS);
        y.w = fmaxf((z.w - m) * r * g[cb + 3] + be[cb + 3], 0.0f);
    }
    *(float4*)(H + (size_t)i * 4) = y;
    if (H2) *(float4*)(H2 + (size_t)i * 4) = y;
}

// ---------------------------------------------------------------------------
// A[dst] += H[src]  (A pre-seeded with H, so result is h + agg). 16 threads/edge.
// h and A are L2-resident (25.6 MB each vs 192 MB L2).
// ---------------------------------------------------------------------------
__global__ void scatter_add(const float* __restrict__ H,
                            const int* __restrict__ src, const int* __restrict__ dst,
                            float* __restrict__ A, int e) {
    const int t = blockIdx.x * blockDim.x + threadIdx.x;
    const int ei = t >> 4;
    if (ei >= e) return;
    const int c4 = (t & 15) * 4;
    const int s = src[ei], d = dst[ei];
    const float4 v = *(const float4*)(H + (size_t)s * HDIM + c4);
    float* o = A + (size_t)d * HDIM + c4;
    atomicAdd(o + 0, v.x); atomicAdd(o + 1, v.y);
    atomicAdd(o + 2, v.z); atomicAdd(o + 3, v.w);
}

// ---------------------------------------------------------------------------
// pooled[batch[node]] += H[node]. 16 threads/node.
// ---------------------------------------------------------------------------
__global__ void pool_sum(const float* __restrict__ H, const int* __restrict__ batch,
                         float* __restrict__ pooled, int n) {
    const int t = blockIdx.x * blockDim.x + threadIdx.x;
    const int node = t >> 4;
    if (node >= n) return;
    const int c4 = (t & 15) * 4;
    const int gid = batch[node];
    const float4 v = *(const float4*)(H + (size_t)node * HDIM + c4);
    float* o = pooled + (size_t)gid * HDIM + c4;
    atomicAdd(o + 0, v.x); atomicAdd(o + 1, v.y);
    atomicAdd(o + 2, v.z); atomicAdd(o + 3, v.w);
}

// ---------------------------------------------------------------------------
// out[g,t] += pooled[g,:] @ W[:,t] + counts[g]*b[t]
// ---------------------------------------------------------------------------
__global__ void out_accum(const float* __restrict__ pooled, const float* __restrict__ W,
                          const float* __restrict__ b, const int* __restrict__ counts,
                          float* __restrict__ out) {
    const int t = blockIdx.x * blockDim.x + threadIdx.x;
    if (t >= GRAPHS * TDIM) return;
    const int g = t / TDIM, c = t % TDIM;
    float acc = (float)counts[g] * b[c];
    const float* p = pooled + (size_t)g * HDIM;
#pragma unroll 8
    for (int f = 0; f < HDIM; ++f) acc += p[f] * W[f * TDIM + c];
    out[t] += acc;
}

__global__ void count_nodes(const int* __restrict__ batch, int* __restrict__ counts, int n) {
    const int i = blockIdx.x * blockDim.x + threadIdx.x;
    if (i < n) atomicAdd(&counts[batch[i]], 1);
}

__global__ void zero_u32(unsigned* __restrict__ p, int n) {
    const int i = blockIdx.x * blockDim.x + threadIdx.x;
    if (i < n) p[i] = 0u;
}

// ---------------------------------------------------------------------------
struct MlpP { const float *W1,*b1,*g1,*be1,*W2,*b2,*g2,*be2; };

extern "C" void kernel_launch(void* const* d_in, const int* in_sizes, int n_in,
                              void* d_out, int out_size, void* d_ws, size_t ws_size,
                              hipStream_t stream) {
    (void)in_sizes; (void)n_in; (void)out_size; (void)ws_size;

    const float* x     = (const float*)d_in[0];            // [N,128]
    const int*   esrc  = (const int*)d_in[1];              // edge_index[0]
    const int*   edst  = esrc + EDGES;                     // edge_index[1]
    const int*   batch = (const int*)d_in[2];              // [N]

    int p = 3;
    auto nextf = [&]() { return (const float*)d_in[p++]; };
    auto mlp = [&]() { MlpP m; m.W1=nextf(); m.b1=nextf(); m.g1=nextf(); m.be1=nextf();
                              m.W2=nextf(); m.b2=nextf(); m.g2=nextf(); m.be2=nextf(); return m; };
    MlpP first = mlp();
    MlpP convs[4]; for (int l = 0; l < 4; ++l) convs[l] = mlp();
    const float* linW[5]; const float* linb[5];
    for (int l = 0; l < 5; ++l) { linW[l] = nextf(); linb[l] = nextf(); }

    float* out = (float*)d_out;

    // workspace layout
    char* ws = (char*)d_ws;
    const size_t NH = (size_t)NODES * HDIM * sizeof(float);     // 25.6 MB
    float* buf0   = (float*)ws;               ws += NH;          // GEMM out (Z)
    float* buf1   = (float*)ws;               ws += NH;          // H
    float* buf2   = (float*)ws;               ws += NH;          // agg seed / MLP mid
    float* stats  = (float*)ws;               ws += 256 * sizeof(float);
    float* pooled = (float*)ws;               ws += (size_t)GRAPHS * HDIM * sizeof(float);
    int*   counts = (int*)ws;                 ws += GRAPHS * sizeof(int);

    const int TB = 256;
    const int gGemm   = (NODES + 127) / 128;                    // 8 waves x 16 rows
    const int gElem4  = (NODES * 16 + TB - 1) / TB;             // float4-vectorized BN
    const int gEdge   = (EDGES * 16 + TB - 1) / TB;
    const int gNode16 = (NODES * 16 + TB - 1) / TB;
    const int gOut    = (GRAPHS * TDIM + TB - 1) / TB;

    // init: out, counts
    zero_u32<<<(GRAPHS*TDIM + TB - 1)/TB, TB, 0, stream>>>((unsigned*)out, GRAPHS*TDIM);
    zero_u32<<<(GRAPHS + TB - 1)/TB, TB, 0, stream>>>((unsigned*)counts, GRAPHS);
    count_nodes<<<(NODES + TB - 1)/TB, TB, 0, stream>>>(batch, counts, NODES);

    // ---- first MLP: x[N,128] -> h[N,64] ----
    gemm_bias_wmma<<<gGemm, TB, 0, stream>>>(x, first.W1, first.b1, buf0, NODES, FIN);
    zero_u32<<<1, 128, 0, stream>>>((unsigned*)stats, 128);
    col_stats<<<256, TB, 0, stream>>>(buf0, stats, NODES);
    bn_relu<<<gElem4, TB, 0, stream>>>(buf0, stats, first.g1, first.be1, buf1, nullptr, NODES);

    gemm_bias_wmma<<<gGemm, TB, 0, stream>>>(buf1, first.W2, first.b2, buf0, NODES, HDIM);
    zero_u32<<<1, 128, 0, stream>>>((unsigned*)stats, 128);
    col_stats<<<256, TB, 0, stream>>>(buf0, stats, NODES);
    bn_relu<<<gElem4, TB, 0, stream>>>(buf0, stats, first.g2, first.be2, buf1, buf2, NODES);
    // buf1 = h, buf2 = h (agg seed)

    // ---- layer-0 readout ----
    zero_u32<<<(GRAPHS*HDIM + TB - 1)/TB, TB, 0, stream>>>((unsigned*)pooled, GRAPHS*HDIM);
    pool_sum<<<gNode16, TB, 0, stream>>>(buf1, batch, pooled, NODES);
    out_accum<<<gOut, TB, 0, stream>>>(pooled, linW[0], linb[0], counts, out);

    // ---- 4 GINConv layers ----
    for (int l = 0; l < 4; ++l) {
        // buf2 = h + segment_sum(h[src], dst)
        scatter_add<<<gEdge, TB, 0, stream>>>(buf1, esrc, edst, buf2, EDGES);

        gemm_bias_wmma<<<gGemm, TB, 0, stream>>>(buf2, convs[l].W1, convs[l].b1, buf0, NODES, HDIM);
        zero_u32<<<1, 128, 0, stream>>>((unsigned*)stats, 128);
        col_stats<<<256, TB, 0, stream>>>(buf0, stats, NODES);
        bn_relu<<<gElem4, TB, 0, stream>>>(buf0, stats, convs[l].g1, convs[l].be1, buf2, nullptr, NODES);

        gemm_bias_wmma<<<gGemm, TB, 0, stream>>>(buf2, convs[l].W2, convs[l].b2, buf0, NODES, HDIM);
        zero_u32<<<1, 128, 0, stream>>>((unsigned*)stats, 128);
        col_stats<<<256, TB, 0, stream>>>(buf0, stats, NODES);
        bn_relu<<<gElem4, TB, 0, stream>>>(buf0, stats, convs[l].g2, convs[l].be2, buf1, buf2, NODES);

        zero_u32<<<(GRAPHS*HDIM + TB - 1)/TB, TB, 0, stream>>>((unsigned*)pooled, GRAPHS*HDIM);
        pool_sum<<<gNode16, TB, 0, stream>>>(buf1, batch, pooled, NODES);
        out_accum<<<gOut, TB, 0, stream>>>(pooled, linW[l + 1], linb[l + 1], counts, out);
    }
}